// LocSE_65197603553785
// MI455X (gfx1250) — compile-verified
//
#include <hip/hip_runtime.h>

// LocSE kNN (K=16) + positional encoding for 16384 points, built around
// V_WMMA_F32_16X16X4_F32 for the pairwise score matrix.
//
// score[i][j] = sq_j - 2*dot(i,j)  (== d2[i][j] - sq_i, ordering-preserving)
// A row i = [x_i, y_i, z_i, 1]  (16x4, f32)
// B col j = [-2x_j, -2y_j, -2z_j, sq_j]  (4x16, f32)

typedef __attribute__((ext_vector_type(2))) float v2f;
typedef __attribute__((ext_vector_type(8))) float v8f;

#define N_PTS 16384
#define KNN   16

// Prepass: pack candidate data so the B operand is one b128 load per lane.
__global__ __launch_bounds__(256) void locse_pack(const float* __restrict__ coords,
                                                  float4* __restrict__ pack) {
  int i = blockIdx.x * blockDim.x + threadIdx.x;
  if (i < N_PTS) {
    float x = coords[3 * i + 0];
    float y = coords[3 * i + 1];
    float z = coords[3 * i + 2];
    pack[i] = make_float4(-2.0f * x, -2.0f * y, -2.0f * z, x * x + y * y + z * z);
  }
}

__global__ __launch_bounds__(256) void locse_knn(const float* __restrict__ coords,
                                                 const float4* __restrict__ pack,
                                                 float* __restrict__ out) {
  // Per-wave staging tile: 32 rows x 16 cols, row stride 18 floats.
  // Stores: halves hit banks {18v..18v+15} vs {18v+16..18v+31} mod 64 -> conflict-free.
  // Loads (b64 per lane, row = lane): 18*dlane % 64 == 0 only at dlane=32 -> conflict-free.
  __shared__ float lds[8][32][18];

  const int lane = threadIdx.x & 31;
  const int warp = threadIdx.x >> 5;
  const int hi   = lane >> 4;   // which K-half this lane holds for A/B operands
  const int col  = lane & 15;
  const int i0   = (blockIdx.x * 8 + warp) * 32;  // first of 32 query rows for this wave

  // ---- Build the two A tiles (rows i0..i0+15 and i0+16..i0+31), resident all scan ----
  // lanes 0-15 : v0 = K0 = x, v1 = K1 = y
  // lanes 16-31: v0 = K2 = z, v1 = K3 = 1
  int qa = i0 + col;
  float ax = coords[3 * qa + 0], ay = coords[3 * qa + 1], az = coords[3 * qa + 2];
  v2f a0;
  a0.x = hi ? az : ax;
  a0.y = hi ? 1.0f : ay;

  int qb = qa + 16;
  float bx_ = coords[3 * qb + 0], by_ = coords[3 * qb + 1], bz_ = coords[3 * qb + 2];
  v2f a1;
  a1.x = hi ? bz_ : bx_;
  a1.y = hi ? 1.0f : by_;

  // ---- This lane owns query row i = i0 + lane for selection & output ----
  const int i = i0 + lane;
  const float cx = coords[3 * i + 0];
  const float cy = coords[3 * i + 1];
  const float cz = coords[3 * i + 2];
  const float sqi = cx * cx + cy * cy + cz * cz;

  // Register-resident sorted top-16 (ascending score). Strict '<' keeps the
  // earliest-scanned (lowest) index on ties, matching jax.lax.top_k.
  float bd[KNN];
  int   bi[KNN];
#pragma unroll
  for (int u = 0; u < KNN; ++u) { bd[u] = 3.0e38f; bi[u] = 0; }

  auto ins = [&](float s, int j) {
    if (s < bd[KNN - 1]) {
#pragma unroll
      for (int u = KNN - 1; u >= 1; --u) {
        bool shiftc = bd[u - 1] > s;   // element above insertion point: shift down
        bool herec  = bd[u] > s;       // insertion point itself
        float nv = shiftc ? bd[u - 1] : (herec ? s : bd[u]);
        int   ni = shiftc ? bi[u - 1] : (herec ? j : bi[u]);
        bd[u] = nv;
        bi[u] = ni;
      }
      if (bd[0] > s) { bd[0] = s; bi[0] = j; }
    }
  };

  const v8f czero = {};

  for (int j0 = 0; j0 < N_PTS; j0 += 16) {
    // B tile (4x16): lanes 0-15: v0=-2x, v1=-2y ; lanes 16-31: v0=-2z, v1=sq
    float4 w = pack[j0 + col];
    v2f b;
    b.x = hi ? w.z : w.x;
    b.y = hi ? w.w : w.y;

    // score tiles: rows i0..i0+15 (c0) and i0+16..i0+31 (c1)
    v8f c0 = __builtin_amdgcn_wmma_f32_16x16x4_f32(false, a0, false, b, (short)0,
                                                   czero, false, false);
    v8f c1 = __builtin_amdgcn_wmma_f32_16x16x4_f32(false, a1, false, b, (short)0,
                                                   czero, false, false);

    // C layout: VGPR v holds M=v (lanes 0-15) / M=v+8 (lanes 16-31), N = lane&15.
    const int rbase = hi ? 8 : 0;
#pragma unroll
    for (int v = 0; v < 8; ++v) {
      lds[warp][rbase + v][col]      = c0[v];
      lds[warp][16 + rbase + v][col] = c1[v];
    }
    // Same-wave RAW through LDS: DS ops are in-order per wave; wait for stores.
    asm volatile("s_wait_dscnt 0" ::: "memory");

    // Each lane reads its own row (16 candidate scores) and merges into top-16.
    const float2* myrow = (const float2*)(&lds[warp][lane][0]);
#pragma unroll
    for (int p = 0; p < 8; ++p) {
      float2 sv = myrow[p];
      ins(sv.x, j0 + 2 * p);
      ins(sv.y, j0 + 2 * p + 1);
    }
    // Keep next iteration's LDS stores from being hoisted above these loads.
    asm volatile("" ::: "memory");
  }

  // ---- Emit pos_enc = [center(3) | neighbor(3) | center-neighbor(3) | dist] ----
  float* op = out + (size_t)i * KNN * 10;
#pragma unroll
  for (int u = 0; u < KNN; ++u) {
    int j = bi[u];
    float nx = coords[3 * j + 0];
    float ny = coords[3 * j + 1];
    float nz = coords[3 * j + 2];
    float d2 = bd[u] + sqi;            // restore row-constant; self -> exactly 0
    float d  = sqrtf(fmaxf(d2, 0.0f));
    op[0] = cx;  op[1] = cy;  op[2] = cz;
    op[3] = nx;  op[4] = ny;  op[5] = nz;
    op[6] = cx - nx;  op[7] = cy - ny;  op[8] = cz - nz;
    op[9] = d;
    op += 10;
  }
}

extern "C" void kernel_launch(void* const* d_in, const int* in_sizes, int n_in,
                              void* d_out, int out_size, void* d_ws, size_t ws_size,
                              hipStream_t stream) {
  const float* coords = (const float*)d_in[0];
  // d_in[1] (features) is unused by the reference pos_enc computation.
  float4* pack = (float4*)d_ws;      // 16384 * 16 B = 256 KB scratch
  float*  out  = (float*)d_out;

  locse_pack<<<N_PTS / 256, 256, 0, stream>>>(coords, pack);
  locse_knn<<<N_PTS / 256, 256, 0, stream>>>(coords, pack, out);
}